// GINConv_4861902979731
// MI455X (gfx1250) — compile-verified
//
#include <hip/hip_runtime.h>

// GIN fused aggregation + GEMM for MI455X (gfx1250, wave32).
// out[r,:] = (sum_{e in row r} X[col(e),:]) @ W      (all fp32)
//
// Per-wave tile: M=16 nodes, K=128, N=128, using V_WMMA_F32_16X16X4_F32.
// W is staged TRANSPOSED in LDS so each B fragment {W[k][n], W[k+1][n]} is a
// single 8B-aligned ds_load_b64 into an even VGPR pair (no repack moves).

typedef float v2f __attribute__((ext_vector_type(2)));
typedef float v8f __attribute__((ext_vector_type(8)));

#define N_NODES_C 100000
#define DEG_C     16
#define DDIM      128                    // D_IN == D_OUT
#define TILE_M    16
#define N_TILES   (N_NODES_C / TILE_M)   // 6250 (exact)
#define WAVES_PB  8
#define THREADS   (WAVES_PB * 32)
#define NB_TILES  (DDIM / 16)            // 8
#define WTS       132                    // padded LDS row stride for W^T (floats, even)
#define AS        132                    // padded LDS row stride for A tile (floats)
#define WT_LDS_F  (DDIM * WTS)           // 16896 floats
#define A_LDS_F   (TILE_M * AS)          // 2112 floats per wave
#define LDS_BYTES ((WT_LDS_F + WAVES_PB * A_LDS_F) * 4)  // 135168 B

__global__ __launch_bounds__(THREADS) void gin_fused_wmma_kernel(
    const float* __restrict__ X,
    const float* __restrict__ W,
    const int*   __restrict__ rowptr,
    const int*   __restrict__ colidx,
    float*       __restrict__ out)
{
    extern __shared__ float smem[];
    float* Wt = smem;                                  // [n=128][k=WTS]  (W transposed)

    // ---- cooperative stage of W^T into LDS (coalesced global reads) ----
    for (int i = threadIdx.x; i < DDIM * DDIM; i += THREADS) {
        const int k = i >> 7;            // W row
        const int n = i & (DDIM - 1);    // W col
        Wt[n * WTS + k] = W[i];
    }
    __syncthreads();

    const int wid  = threadIdx.x >> 5;
    const int lane = threadIdx.x & 31;
    const int tile = blockIdx.x * WAVES_PB + wid;
    if (tile >= N_TILES) return;                       // wave-uniform; EXEC stays all-1s
    const int node_base = tile * TILE_M;

    float* Al = smem + WT_LDS_F + wid * A_LDS_F;       // per-wave [16][AS]

    // ---- aggregation: lane owns feature cols [4*lane, 4*lane+4) ----
    // Two nodes per iteration -> 32 outstanding b128 gathers per lane to
    // hide L2 latency (X is L2-resident: 51MB vs 192MB L2).
    for (int m = 0; m < TILE_M; m += 2) {
        const int node0 = node_base + m;
        const int rs0   = rowptr[node0];               // wave-uniform -> SMEM
        const int rs1   = rowptr[node0 + 1];
        int cols0[DEG_C], cols1[DEG_C];
        #pragma unroll
        for (int e = 0; e < DEG_C; ++e) cols0[e] = colidx[rs0 + e];
        #pragma unroll
        for (int e = 0; e < DEG_C; ++e) cols1[e] = colidx[rs1 + e];

        float4 acc0 = make_float4(0.f, 0.f, 0.f, 0.f);
        float4 acc1 = make_float4(0.f, 0.f, 0.f, 0.f);
        #pragma unroll
        for (int e = 0; e < DEG_C; ++e) {              // independent b128 gathers
            const float4 v0 = *((const float4*)(X + (size_t)cols0[e] * DDIM) + lane);
            const float4 v1 = *((const float4*)(X + (size_t)cols1[e] * DDIM) + lane);
            acc0.x += v0.x; acc0.y += v0.y; acc0.z += v0.z; acc0.w += v0.w;
            acc1.x += v1.x; acc1.y += v1.y; acc1.z += v1.z; acc1.w += v1.w;
        }
        *(float4*)(Al + (m + 0) * AS + 4 * lane) = acc0;  // 16B-aligned (528B pitch)
        *(float4*)(Al + (m + 1) * AS + 4 * lane) = acc1;
    }
    // Same-wave LDS write->read is in-order; compiler inserts waits as needed.

    // ---- GEMM: D[16,128] = A[16,128] x W[128,128] via f32 WMMA 16x16x4 ----
    const int m16   = lane & 15;
    const int half  = lane >> 4;       // 0: lanes 0-15, 1: lanes 16-31
    const int khalf = half * 2;        // K offset per ISA A/B VGPR layout

    v8f c[NB_TILES];
    #pragma unroll
    for (int nb = 0; nb < NB_TILES; ++nb) c[nb] = (v8f){};

    const float* Arow = Al + m16 * AS;                 // A[m16][*]
    const float* Wrow = Wt + m16 * WTS;                // W^T[m16 + nb*16][*]
    #pragma unroll 2
    for (int k = 0; k < DDIM; k += 4) {
        const int k0 = k + khalf;
        // A fragment loaded ONCE per k-step: {A[m][k0], A[m][k0+1]} (ds_load_b64)
        const v2f a = *(const v2f*)(Arow + k0);
        #pragma unroll
        for (int nb = 0; nb < NB_TILES; ++nb) {
            // B fragment: {W[k0][n], W[k0+1][n]} contiguous in W^T -> ds_load_b64
            // (immediate offset nb*8448 + k0*4 < 64KB folds into the DS op)
            const v2f b = *(const v2f*)(Wrow + nb * (16 * WTS) + k0);
            c[nb] = __builtin_amdgcn_wmma_f32_16x16x4_f32(
                        /*neg_a=*/false, a, /*neg_b=*/false, b,
                        /*c_mod=*/(short)0, c[nb], /*reuse_a=*/false, /*reuse_b=*/false);
        }
    }

    // ---- store: C/D layout: VGPR v -> row (v + 8*half), col nb*16 + m16 ----
    float* obase = out + (size_t)(node_base + half * 8) * DDIM + m16;
    #pragma unroll
    for (int nb = 0; nb < NB_TILES; ++nb) {
        float* orow = obase + nb * 16;
        #pragma unroll
        for (int v = 0; v < 8; ++v) orow[(size_t)v * DDIM] = c[nb][v];
    }
}

extern "C" void kernel_launch(void* const* d_in, const int* in_sizes, int n_in,
                              void* d_out, int out_size, void* d_ws, size_t ws_size,
                              hipStream_t stream)
{
    const float* X      = (const float*)d_in[0];
    const float* W      = (const float*)d_in[1];
    const int*   rowptr = (const int*)d_in[2];
    const int*   colidx = (const int*)d_in[3];
    float*       out    = (float*)d_out;

    // Allow >64KB dynamic LDS (WGP supports up to 320KB per workgroup).
    (void)hipFuncSetAttribute((const void*)gin_fused_wmma_kernel,
                              hipFuncAttributeMaxDynamicSharedMemorySize, LDS_BYTES);

    const int grid = (N_TILES + WAVES_PB - 1) / WAVES_PB;  // 782
    gin_fused_wmma_kernel<<<grid, THREADS, LDS_BYTES, stream>>>(X, W, rowptr, colidx, out);
}